// dense_knn_matrix_74002286510477
// MI455X (gfx1250) — compile-verified
//
#include <hip/hip_runtime.h>

typedef __attribute__((ext_vector_type(2))) float v2f;
typedef __attribute__((ext_vector_type(8))) float v8f;

#define B_BATCH 2
#define N_PTS   8192
#define DIMS    64
#define KNN     16

// One wave (32 lanes) owns a 16-row query tile. Per iteration it produces a
// 16x32 distance tile via 32x V_WMMA_F32_16X16X4_F32 (16 K-steps x 2 column
// tiles), writes it to LDS, and each lane maintains a sorted top-16 for one
// (row, column-half) pair. Final cross-half merge via shfl.
__global__ __launch_bounds__(128) void dense_knn_kernel(const float* __restrict__ x,
                                                        int* __restrict__ out) {
    __shared__ float lds_rsq[4][16];
    __shared__ float lds_d[4][16][33];   // [wave][row][col], padded stride 33

    const int lane = threadIdx.x & 31;
    const int wave = threadIdx.x >> 5;
    const int b    = blockIdx.y;
    const int row_base = blockIdx.x * 64 + wave * 16;

    const float* xb = x + (size_t)b * N_PTS * DIMS;

    const int lrow = lane & 15;   // row (A) / column (B) within tile
    const int half = lane >> 4;   // which K-pair this lane holds
    const int koff = half * 2;

    // ---- Load A tile: 16 rows x 64 dims -> 16 x v2f, and row sq-norms ----
    v2f a[16];
    float asq = 0.0f;
    #pragma unroll
    for (int kk = 0; kk < 16; ++kk) {
        const float* p = xb + (size_t)(row_base + lrow) * DIMS + 4 * kk + koff;
        a[kk] = *(const v2f*)p;
        asq += a[kk].x * a[kk].x + a[kk].y * a[kk].y;
    }
    asq += __shfl_xor(asq, 16, 32);
    if (lane < 16) lds_rsq[wave][lane] = asq;
    __syncthreads();

    // ---- Per-lane top-K (sorted ascending) ----
    float best_d[KNN];
    int   best_i[KNN];
    #pragma unroll
    for (int j = 0; j < KNN; ++j) { best_d[j] = 3.4e38f; best_i[j] = 0; }

    for (int col_base = 0; col_base < N_PTS; col_base += 32) {
        v8f c0 = {}; v8f c1 = {};
        float csq0 = 0.0f, csq1 = 0.0f;

        #pragma unroll
        for (int kk = 0; kk < 16; ++kk) {
            const float* p0 = xb + (size_t)(col_base + lrow) * DIMS + 4 * kk + koff;
            const float* p1 = p0 + 16 * DIMS;
            v2f b0 = *(const v2f*)p0;
            v2f b1 = *(const v2f*)p1;
            csq0 += b0.x * b0.x + b0.y * b0.y;
            csq1 += b1.x * b1.x + b1.y * b1.y;
            c0 = __builtin_amdgcn_wmma_f32_16x16x4_f32(false, a[kk], false, b0,
                                                       (short)0, c0, false, false);
            c1 = __builtin_amdgcn_wmma_f32_16x16x4_f32(false, a[kk], false, b1,
                                                       (short)0, c1, false, false);
        }
        csq0 += __shfl_xor(csq0, 16, 32);
        csq1 += __shfl_xor(csq1, 16, 32);

        // dist = ||xi||^2 + ||xj||^2 - 2*dot ; C layout: (m = v+8*half, n = lrow)
        #pragma unroll
        for (int v = 0; v < 8; ++v) {
            int m = v + 8 * half;
            float rsq = lds_rsq[wave][m];
            lds_d[wave][m][lrow]      = rsq + csq0 - 2.0f * c0[v];
            lds_d[wave][m][16 + lrow] = rsq + csq1 - 2.0f * c1[v];
        }
        __syncthreads();

        // lane l scans row (l&15), columns half*16 .. half*16+15 of this chunk
        const int col0 = half * 16;
        #pragma unroll
        for (int n = 0; n < 16; ++n) {
            float d = lds_d[wave][lrow][col0 + n];
            if (d < best_d[KNN - 1]) {
                float cd = d;
                int   ci = col_base + col0 + n;
                #pragma unroll
                for (int j = 0; j < KNN; ++j) {
                    if (cd < best_d[j]) {
                        float td = best_d[j]; best_d[j] = cd; cd = td;
                        int   ti = best_i[j]; best_i[j] = ci; ci = ti;
                    }
                }
            }
        }
        __syncthreads();
    }

    // ---- Merge lane l's list with lane (l+16)'s list (same row) ----
    #pragma unroll
    for (int j = 0; j < KNN; ++j) {
        float od = __shfl(best_d[j], lrow + 16, 32);
        int   oi = __shfl(best_i[j], lrow + 16, 32);
        if (lane < 16 && od < best_d[KNN - 1]) {
            float cd = od; int ci = oi;
            #pragma unroll
            for (int t = 0; t < KNN; ++t) {
                if (cd < best_d[t]) {
                    float td = best_d[t]; best_d[t] = cd; cd = td;
                    int   ti = best_i[t]; best_i[t] = ci; ci = ti;
                }
            }
        }
    }

    // ---- Write output: plane0 = nn_idx, plane1 = center_idx ----
    if (lane < 16) {
        const int row = row_base + lrow;
        const size_t base  = ((size_t)b * N_PTS + row) * KNN;
        const size_t plane = (size_t)B_BATCH * N_PTS * KNN;
        #pragma unroll
        for (int j = 0; j < KNN; ++j) {
            out[base + j]         = best_i[j];
            out[plane + base + j] = row;
        }
    }
}

extern "C" void kernel_launch(void* const* d_in, const int* in_sizes, int n_in,
                              void* d_out, int out_size, void* d_ws, size_t ws_size,
                              hipStream_t stream) {
    (void)in_sizes; (void)n_in; (void)out_size; (void)d_ws; (void)ws_size;
    const float* x = (const float*)d_in[0];
    int* out = (int*)d_out;
    dim3 grid(N_PTS / 64, B_BATCH);
    dense_knn_kernel<<<grid, 128, 0, stream>>>(x, out);
}